// WarpLayer_48490180772278
// MI455X (gfx1250) — compile-verified
//
#include <hip/hip_runtime.h>
#include <hip/hip_bf16.h>
#include <math.h>

// ---------------------------------------------------------------------------
// CDNA5 async global->LDS helpers (gfx1250).  Inline asm: portable across
// ROCm 7.2 / amdgpu-toolchain (bypasses builtin-arity differences).
//   global_load_async_to_lds_b128 vdst(lds byte addr), vaddr(64b), off
// Generic pointers to __shared__ carry the LDS offset in their low 32 bits,
// so (unsigned)(size_t)&lds[i] is the correct VDST value (HW adds LDS_BASE).
// ---------------------------------------------------------------------------
#define ASYNC_LD_B128(ldsOff, gptr)                                          \
  asm volatile("global_load_async_to_lds_b128 %0, %1, off"                   \
               :: "v"(ldsOff), "v"(gptr) : "memory")

#define WAIT_ASYNC(n)                                                        \
  asm volatile("s_wait_asynccnt %0" :: "n"(n) : "memory")

#define TILE 1024   // elements per tile per block (256 thr * 4 elem)
#define BLK  256
#define NSTAT 28    // 4 groups * {cnt, St, Stt, Sx, Stx, Sy, Sty}

// ---------------------------------------------------------------------------
// Phase 0: zero the 28 stat accumulators (+ padding) in workspace.
// ---------------------------------------------------------------------------
__global__ void wl_zero_kernel(float* ws) {
  if (threadIdx.x < 64) ws[threadIdx.x] = 0.0f;
}

// ---------------------------------------------------------------------------
// Phase 1: streaming segmented sufficient statistics.
// Async-DMA double-buffered pipeline: each lane issues 4x b128 async loads
// (x,y,t,b) into its private LDS slots for tile i+1, waits asynccnt<=4 for
// tile i, consumes from LDS with ds_load_b128, accumulates 28 partials.
// ---------------------------------------------------------------------------
__global__ __launch_bounds__(BLK)
void wl_stats_kernel(const float* __restrict__ x, const float* __restrict__ y,
                     const float* __restrict__ t, const int* __restrict__ b,
                     float* __restrict__ stats, int N) {
  __shared__ __align__(16) float lx[2][TILE];
  __shared__ __align__(16) float ly[2][TILE];
  __shared__ __align__(16) float lt[2][TILE];
  __shared__ __align__(16) int   lb[2][TILE];
  __shared__ float red[BLK / 32][NSTAT];

  const int tid = threadIdx.x;
  const int numTiles = N / TILE;

  float acc[NSTAT];
#pragma unroll
  for (int j = 0; j < NSTAT; ++j) acc[j] = 0.0f;

  auto issue = [&](int tileIdx, int bsel) {
    const int e = tileIdx * TILE + tid * 4;
    ASYNC_LD_B128((unsigned)(size_t)&lx[bsel][tid * 4], x + e);
    ASYNC_LD_B128((unsigned)(size_t)&ly[bsel][tid * 4], y + e);
    ASYNC_LD_B128((unsigned)(size_t)&lt[bsel][tid * 4], t + e);
    ASYNC_LD_B128((unsigned)(size_t)&lb[bsel][tid * 4], b + e);
  };

  auto accumulate = [&](float X, float Y, float T, int B_) {
    const float TT = T * T, TX = T * X, TY = T * Y;
#pragma unroll
    for (int g = 0; g < 4; ++g) {
      const float m = (B_ == g) ? 1.0f : 0.0f;
      acc[g * 7 + 0] += m;
      acc[g * 7 + 1] += m * T;
      acc[g * 7 + 2] += m * TT;
      acc[g * 7 + 3] += m * X;
      acc[g * 7 + 4] += m * TX;
      acc[g * 7 + 5] += m * Y;
      acc[g * 7 + 6] += m * TY;
    }
  };

  // --- double-buffered grid-strided tile pipeline ---
  int tile = (int)blockIdx.x;
  int buf = 0;
  if (tile < numTiles) issue(tile, 0);
  while (tile < numTiles) {
    const int next = tile + (int)gridDim.x;
    const bool pf = (next < numTiles);
    if (pf) {
      issue(next, buf ^ 1);
      WAIT_ASYNC(4);   // current tile's 4 loads done; prefetch in flight
    } else {
      WAIT_ASYNC(0);
    }
    const float4 xv = *(const float4*)&lx[buf][tid * 4];
    const float4 yv = *(const float4*)&ly[buf][tid * 4];
    const float4 tv = *(const float4*)&lt[buf][tid * 4];
    const int4   bv = *(const int4*)  &lb[buf][tid * 4];
    accumulate(xv.x, yv.x, tv.x, bv.x);
    accumulate(xv.y, yv.y, tv.y, bv.y);
    accumulate(xv.z, yv.z, tv.z, bv.z);
    accumulate(xv.w, yv.w, tv.w, bv.w);
    buf ^= 1;
    tile = next;
  }

  // --- tail (N not multiple of TILE): plain scalar loads ---
  for (int i = numTiles * TILE + (int)blockIdx.x * BLK + tid; i < N;
       i += (int)gridDim.x * BLK) {
    accumulate(x[i], y[i], t[i], b[i]);
  }

  // --- wave32 reduction, then cross-wave via LDS, then global atomics ---
  const int lane = tid & 31;
  const int wave = tid >> 5;
#pragma unroll
  for (int j = 0; j < NSTAT; ++j) {
    float v = acc[j];
    v += __shfl_down(v, 16, 32);
    v += __shfl_down(v, 8, 32);
    v += __shfl_down(v, 4, 32);
    v += __shfl_down(v, 2, 32);
    v += __shfl_down(v, 1, 32);
    if (lane == 0) red[wave][j] = v;
  }
  __syncthreads();
  if (tid < NSTAT) {
    float s = 0.0f;
#pragma unroll
    for (int w = 0; w < BLK / 32; ++w) s += red[w][tid];
    atomicAdd(&stats[tid], s);
  }
}

// ---------------------------------------------------------------------------
// Phase 2: Adam.  Loss is quadratic in p, so grad = A*p + C with fixed A,C.
// The 8 parameters are fully decoupled: one thread each runs 20 steps.
// ---------------------------------------------------------------------------
__global__ void wl_adam_kernel(const float* __restrict__ stats,
                               const float* __restrict__ parms0,
                               float* __restrict__ pOut) {
  const int i = threadIdx.x;
  if (i >= 8) return;
  const int g = i >> 1, c = i & 1;
  const float cnt = stats[g * 7 + 0];
  const float T   = stats[g * 7 + 1];
  const float Stt = stats[g * 7 + 2];
  const float Sv  = c ? stats[g * 7 + 5] : stats[g * 7 + 3];  // Σy or Σx
  const float Stv = c ? stats[g * 7 + 6] : stats[g * 7 + 4];  // Σty or Σtx
  const float inv_d = 1.0f / (cnt - 1.0f);
  const float A =  2.0f * (Stt - T * T / cnt) * inv_d;
  const float C = -2.0f * (Stv - Sv * T / cnt) * inv_d;

  float p = parms0[i];
  float m = 0.0f, v = 0.0f, b1p = 1.0f, b2p = 1.0f;
#pragma unroll
  for (int s = 0; s < 20; ++s) {
    const float gr = A * p + C;
    m = 0.9f * m + 0.1f * gr;
    v = 0.999f * v + 0.001f * gr * gr;
    b1p *= 0.9f;
    b2p *= 0.999f;
    const float mh = m / (1.0f - b1p);
    const float vh = v / (1.0f - b2p);
    p -= 0.01f * mh / (sqrtf(vh) + 1e-8f);
  }
  pOut[i] = p;
}

// ---------------------------------------------------------------------------
// Phase 3: final warp + out-of-sensor zeroing.  Pure streaming, float4 I/O.
// ---------------------------------------------------------------------------
__global__ __launch_bounds__(BLK)
void wl_warp_kernel(const float* __restrict__ x, const float* __restrict__ y,
                    const float* __restrict__ t, const int* __restrict__ b,
                    const float* __restrict__ p, float* __restrict__ outx,
                    float* __restrict__ outy, int N) {
  const float p0x = p[0], p0y = p[1], p1x = p[2], p1y = p[3];
  const float p2x = p[4], p2y = p[5], p3x = p[6], p3y = p[7];

  auto warp1 = [&](float X, float Y, float T, int B_, float& ox, float& oy) {
    const float px = (B_ == 0) ? p0x : (B_ == 1) ? p1x : (B_ == 2) ? p2x : p3x;
    const float py = (B_ == 0) ? p0y : (B_ == 1) ? p1y : (B_ == 2) ? p2y : p3y;
    float xo = X - T * px;
    float yo = Y - T * py;
    ox = (xo < 0.0f || xo >= 239.0f) ? 0.0f : xo;
    oy = (yo < 0.0f || yo >= 179.0f) ? 0.0f : yo;
  };

  const int nv = N >> 2;
  for (int i = (int)blockIdx.x * BLK + (int)threadIdx.x; i < nv;
       i += (int)gridDim.x * BLK) {
    const float4 xv = ((const float4*)x)[i];
    const float4 yv = ((const float4*)y)[i];
    const float4 tv = ((const float4*)t)[i];
    const int4   bv = ((const int4*)b)[i];
    float4 ox, oy;
    warp1(xv.x, yv.x, tv.x, bv.x, ox.x, oy.x);
    warp1(xv.y, yv.y, tv.y, bv.y, ox.y, oy.y);
    warp1(xv.z, yv.z, tv.z, bv.z, ox.z, oy.z);
    warp1(xv.w, yv.w, tv.w, bv.w, ox.w, oy.w);
    ((float4*)outx)[i] = ox;
    ((float4*)outy)[i] = oy;
  }
  // tail
  for (int i = nv * 4 + (int)blockIdx.x * BLK + (int)threadIdx.x; i < N;
       i += (int)gridDim.x * BLK) {
    float ox, oy;
    warp1(x[i], y[i], t[i], b[i], ox, oy);
    outx[i] = ox;
    outy[i] = oy;
  }
}

// ---------------------------------------------------------------------------
extern "C" void kernel_launch(void* const* d_in, const int* in_sizes, int n_in,
                              void* d_out, int out_size, void* d_ws,
                              size_t ws_size, hipStream_t stream) {
  const float* x = (const float*)d_in[0];
  const float* y = (const float*)d_in[1];
  const float* t = (const float*)d_in[2];
  const int*   b = (const int*)d_in[3];
  const float* parms = (const float*)d_in[4];
  const int N = in_sizes[0];

  float* ws    = (float*)d_ws;
  float* stats = ws;        // [0..27]  segmented sums
  float* pfin  = ws + 32;   // [32..39] calibrated parameters

  float* outx = (float*)d_out;
  float* outy = (float*)d_out + N;

  wl_zero_kernel<<<1, 64, 0, stream>>>(ws);
  wl_stats_kernel<<<1024, BLK, 0, stream>>>(x, y, t, b, stats, N);
  wl_adam_kernel<<<1, 32, 0, stream>>>(stats, parms, pfin);
  wl_warp_kernel<<<2048, BLK, 0, stream>>>(x, y, t, b, pfin, outx, outy, N);
}